// ContrastiveLossReRank_26843545600182
// MI455X (gfx1250) — compile-verified
//
#include <hip/hip_runtime.h>

// ---------------------------------------------------------------------------
// ContrastiveLossReRank for MI455X (gfx1250, wave32)
//   N=256, C=1024, K=2048, T=0.07
//   HBM-bound: neg[] = 2.15 GB read once -> ~92us floor @ 23.3 TB/s.
//   Pipeline: GLOBAL_LOAD_ASYNC_TO_LDS_B128 (ASYNCcnt, non-temporal,
//   double-buffered per-wave LDS tiles, fully-coalesced 128B/row chunks)
//   -> V_WMMA_F32_16X16X4_F32 (A = broadcast q chunk, B = 16-row neg tile).
// ---------------------------------------------------------------------------

typedef float v2f __attribute__((ext_vector_type(2)));
typedef float v8f __attribute__((ext_vector_type(8)));

#define N_ 256
#define C_ 1024
#define K_ 2048
#define INV_T 14.2857142857142857f /* 1/0.07 */

// Per-wave streaming geometry: 8 waves, each owns 256 rows = 8 sets of 32 rows.
// Each chunk = 32 floats (128 B) per row for 32 rows = 4 KB staged in LDS.
#define CHUNK 32
#define NSETS 8
#define NCH   32            /* C_ / CHUNK */
#define SLOTS (NSETS * NCH) /* 256 */

__global__ __launch_bounds__(256) void cl_row_kernel(
    const float* __restrict__ q, const float* __restrict__ kk,
    const float* __restrict__ neg, float* __restrict__ row_loss)
{
    __shared__ float s_q[C_];                       // 4 KB
    __shared__ float s_logits[K_ + 1];              // 8 KB
    __shared__ float s_red[256];                    // 1 KB
    __shared__ __align__(16) float s_stage[8][2][32][CHUNK]; // 64 KB, per-wave dbl buf

    const int t = threadIdx.x;
    const int n = blockIdx.x;
    const float* qrow = q  + (size_t)n * C_;
    const float* krow = kk + (size_t)n * C_;

    // ---- stage q[n] into LDS, and l_pos = q.k via deterministic tree ----
    float p = 0.f;
    #pragma unroll
    for (int i = t; i < C_; i += 256) {
        float qv = qrow[i];
        s_q[i] = qv;
        p += qv * krow[i];
    }
    s_red[t] = p;
    __syncthreads();
    for (int s = 128; s > 0; s >>= 1) {
        if (t < s) s_red[t] += s_red[t + s];
        __syncthreads();
    }
    if (t == 0) s_logits[0] = s_red[0] * INV_T;
    __syncthreads();

    // ---- l_neg via async-LDS-staged WMMA stream ----
    const int wave = t >> 5;
    const int lane = t & 31;
    const int col  = lane & 15;        // B column = row within 16-row group
    const int koff = (lane >> 4) * 2;  // K-pair: lanes<16 -> {0,1}, >=16 -> {2,3}
    const int r4   = lane >> 3;        // 0..3 : row-within-4 for coalesced loads
    const int i8   = lane & 7;         // 0..7 : 16B segment within 128B row chunk

    const float* negn = neg + (size_t)n * K_ * C_;
    float* stg = &s_stage[wave][0][0][0];          // [2][32][CHUNK]

    // Issue one chunk-slot: 8 async non-temporal b128 loads, each covering
    // 4 full rows' 128B (fully coalesced, whole cachelines).
    auto issue = [&](int slot) {
        const int buf = slot & 1;
        const int set = slot >> 5;        // which 32-row set
        const int ch  = slot & 31;        // which 128B column chunk
        const int rowbase = wave * 32 + set * 256;
        const float* g0 = negn + (size_t)(rowbase + r4) * C_ + ch * CHUNK + i8 * 4;
        const float* l0f = stg + buf * (32 * CHUNK) + r4 * CHUNK + i8 * 4;
        unsigned int l0 = (unsigned int)(unsigned long long)l0f; // LDS byte offset
        #pragma unroll
        for (int rr = 0; rr < 8; ++rr) {
            const float* gp = g0 + (size_t)rr * 4 * C_;
            unsigned int lp = l0 + rr * 4 * CHUNK * 4;
            asm volatile("global_load_async_to_lds_b128 %0, %1, off th:TH_LOAD_NT"
                         :: "v"(lp), "v"(gp) : "memory");
        }
    };

    issue(0);
    int slot = 0;
    #pragma unroll 1
    for (int set = 0; set < NSETS; ++set) {
        v8f acc0 = {};
        v8f acc1 = {};
        #pragma unroll 1
        for (int ch = 0; ch < NCH; ++ch, ++slot) {
            const int buf = slot & 1;
            if (slot + 1 < SLOTS) {
                issue(slot + 1);
                asm volatile("s_wait_asynccnt 0x8" ::: "memory"); // slot's 8 landed
            } else {
                asm volatile("s_wait_asynccnt 0x0" ::: "memory");
            }

            const float* bq = s_q + ch * CHUNK;
            const float* b0 = stg + buf * (32 * CHUNK) + col * CHUNK;
            const float* b1 = b0 + 16 * CHUNK;
            #pragma unroll
            for (int s8 = 0; s8 < 8; ++s8) {
                v2f a;                                   // A[m,k] = q[cc+k] (all m)
                a.x = bq[s8 * 4 + koff];
                a.y = bq[s8 * 4 + koff + 1];
                v2f vb0 = *(const v2f*)(b0 + s8 * 4 + koff);
                v2f vb1 = *(const v2f*)(b1 + s8 * 4 + koff);
                acc0 = __builtin_amdgcn_wmma_f32_16x16x4_f32(
                           false, a, false, vb0, (short)0, acc0, false, false);
                acc1 = __builtin_amdgcn_wmma_f32_16x16x4_f32(
                           false, a, false, vb1, (short)0, acc1, false, false);
            }
        }
        // D row M=0 lives in component 0; lanes 0..15 hold columns 0..15.
        const int rowbase = wave * 32 + set * 256;
        if (lane < 16) {
            s_logits[1 + rowbase + lane]      = acc0[0] * INV_T;
            s_logits[1 + rowbase + 16 + lane] = acc1[0] * INV_T;
        }
    }
    __syncthreads();

    // ---- per-row softmax cross-entropy: lse(logits) - logits[0] ----
    float lmax = -3.402823466e+38f;
    for (int i = t; i < K_ + 1; i += 256) lmax = fmaxf(lmax, s_logits[i]);
    s_red[t] = lmax;
    __syncthreads();
    for (int s = 128; s > 0; s >>= 1) {
        if (t < s) s_red[t] = fmaxf(s_red[t], s_red[t + s]);
        __syncthreads();
    }
    const float m = s_red[0];
    __syncthreads();

    float lsum = 0.f;
    for (int i = t; i < K_ + 1; i += 256) lsum += __expf(s_logits[i] - m);
    s_red[t] = lsum;
    __syncthreads();
    for (int s = 128; s > 0; s >>= 1) {
        if (t < s) s_red[t] += s_red[t + s];
        __syncthreads();
    }
    if (t == 0) row_loss[n] = __logf(s_red[0]) + m - s_logits[0];
}

__global__ __launch_bounds__(256) void cl_reduce_kernel(
    const float* __restrict__ row_loss, float* __restrict__ out)
{
    __shared__ float s_red[256];
    const int t = threadIdx.x;
    s_red[t] = row_loss[t];
    __syncthreads();
    for (int s = 128; s > 0; s >>= 1) {
        if (t < s) s_red[t] += s_red[t + s];
        __syncthreads();
    }
    if (t == 0) out[0] = s_red[0] * (1.0f / (float)N_);
}

extern "C" void kernel_launch(void* const* d_in, const int* in_sizes, int n_in,
                              void* d_out, int out_size, void* d_ws, size_t ws_size,
                              hipStream_t stream) {
    const float* q   = (const float*)d_in[0];
    const float* k   = (const float*)d_in[1];
    const float* neg = (const float*)d_in[2];
    float* row_loss  = (float*)d_ws;            // 256 floats of scratch

    cl_row_kernel<<<N_, 256, 0, stream>>>(q, k, neg, row_loss);
    cl_reduce_kernel<<<1, 256, 0, stream>>>(row_loss, (float*)d_out);
}